// SupervisedContrastiveLoss_78228534329348
// MI455X (gfx1250) — compile-verified
//
#include <hip/hip_runtime.h>
#include <hip/hip_bf16.h>

typedef float v2f __attribute__((ext_vector_type(2)));
typedef float v8f __attribute__((ext_vector_type(8)));

#define N_ROWS 8192
#define DIM    128

// ---------------------------------------------------------------------------
// K1: L2-normalize each row (one 128-thread block per row)
// ---------------------------------------------------------------------------
__global__ void __launch_bounds__(128)
scl_normalize(const float* __restrict__ emb, float* __restrict__ enorm) {
    const int row = blockIdx.x;
    const int tid = threadIdx.x;           // 0..127
    float x = emb[row * DIM + tid];
    float s = x * x;
    #pragma unroll
    for (int o = 16; o >= 1; o >>= 1) s += __shfl_xor(s, o, 32);
    __shared__ float part[4];
    __shared__ float inv;
    if ((tid & 31) == 0) part[tid >> 5] = s;
    __syncthreads();
    if (tid == 0) {
        float t = part[0] + part[1] + part[2] + part[3];
        inv = 1.0f / sqrtf(t);
    }
    __syncthreads();
    enorm[row * DIM + tid] = x * inv;
}

// ---------------------------------------------------------------------------
// K2: fused sim-row reductions via f32 WMMA, double-buffered B fragments.
//   Block b owns rows [16b, 16b+16). Wave w (0..7) sweeps columns
//   [1024w, 1024w+1024) in 16-wide tiles. While tile t runs its 32 chained
//   v_wmma_f32_16x16x4_f32 + exp epilogue, tile t+1's 32 global_load_b64
//   are in flight into the other buffer.
// ---------------------------------------------------------------------------
__global__ void __launch_bounds__(256)
scl_sim_rows(const float* __restrict__ enorm, const int* __restrict__ labels,
             float* __restrict__ denom_out, float* __restrict__ pos_out) {
    const int i0   = blockIdx.x * 16;
    const int wave = threadIdx.x >> 5;
    const int lane = threadIdx.x & 31;
    const int l16  = lane & 15;
    const int hi   = lane >> 4;

    // Preload A fragments (rows i0..i0+15, all K). ISA layout: lane = M%16,
    // VGPR pair holds K = 4*kt + (hi?2:0), +1.
    const float* aptr = enorm + (size_t)(i0 + l16) * DIM + hi * 2;
    v2f afrag[32];
    #pragma unroll
    for (int kt = 0; kt < 32; ++kt)
        afrag[kt] = *(const v2f*)(aptr + kt * 4);

    // Row labels matching the C layout (lanes 0-15 -> M=v, lanes 16-31 -> M=v+8)
    int labrow[8];
    #pragma unroll
    for (int v = 0; v < 8; ++v) labrow[v] = labels[i0 + v + hi * 8];

    float dacc[8], pacc[8];
    #pragma unroll
    for (int v = 0; v < 8; ++v) { dacc[v] = 0.f; pacc[v] = 0.f; }

    // B base for this wave's column chunk; tile t lives at +t*16*DIM floats.
    const float* bbase = enorm + (size_t)(wave * 1024 + l16) * DIM + hi * 2;

    auto loadB = [&](v2f* dst, int tile) {
        const float* bp = bbase + (size_t)tile * 16 * DIM;
        #pragma unroll
        for (int kt = 0; kt < 32; ++kt) dst[kt] = *(const v2f*)(bp + kt * 4);
    };

    auto tileCompute = [&](const v2f* bfr, int t) {
        v8f c = {};
        #pragma unroll
        for (int kt = 0; kt < 32; ++kt) {
            c = __builtin_amdgcn_wmma_f32_16x16x4_f32(
                    /*neg_a=*/false, afrag[kt], /*neg_b=*/false, bfr[kt],
                    /*c_mod=*/(short)0, c, /*reuse_a=*/false, /*reuse_b=*/false);
        }
        const int col    = wave * 1024 + t * 16 + l16;
        const int labcol = labels[col];
        #pragma unroll
        for (int v = 0; v < 8; ++v) {
            const int   row  = i0 + v + hi * 8;
            const float s    = c[v];
            const bool  diag = (row == col);
            dacc[v] += diag ? 0.f : __expf(s);
            pacc[v] += (!diag && (labrow[v] == labcol)) ? s : 0.f;
        }
    };

    v2f b0[32], b1[32];
    loadB(b0, 0);
    for (int tt = 0; tt < 64; tt += 2) {
        loadB(b1, tt + 1);             // prefetch odd tile while computing even
        tileCompute(b0, tt);
        loadB(b0, (tt + 2) & 63);      // branch-free wrap prefetch of next even
        tileCompute(b1, tt + 1);
    }

    // Butterfly-reduce across the 16 lanes sharing each row (per C-layout half)
    #pragma unroll
    for (int v = 0; v < 8; ++v) {
        float d = dacc[v], p = pacc[v];
        #pragma unroll
        for (int o = 8; o >= 1; o >>= 1) {
            d += __shfl_xor(d, o, 16);
            p += __shfl_xor(p, o, 16);
        }
        dacc[v] = d; pacc[v] = p;
    }

    __shared__ float lds_d[8][16];
    __shared__ float lds_p[8][16];
    if (l16 == 0) {
        #pragma unroll
        for (int v = 0; v < 8; ++v) {
            lds_d[wave][hi * 8 + v] = dacc[v];
            lds_p[wave][hi * 8 + v] = pacc[v];
        }
    }
    __syncthreads();
    if (threadIdx.x < 16) {
        float d = 0.f, p = 0.f;
        #pragma unroll
        for (int w = 0; w < 8; ++w) { d += lds_d[w][threadIdx.x]; p += lds_p[w][threadIdx.x]; }
        denom_out[i0 + threadIdx.x] = d;
        pos_out[i0 + threadIdx.x]   = p;
    }
}

// ---------------------------------------------------------------------------
// K3: deterministic final reduction (single block).
// ---------------------------------------------------------------------------
__global__ void __launch_bounds__(1024)
scl_finalize(const int* __restrict__ labels, const float* __restrict__ denom,
             const float* __restrict__ pos, float* __restrict__ out) {
    const int tid = threadIdx.x;
    __shared__ int   ci[32];
    __shared__ float fa[32], fb[32];
    __shared__ int   c0s;

    int c = 0;
    for (int i = tid; i < N_ROWS; i += 1024) c += (labels[i] == 0) ? 1 : 0;
    #pragma unroll
    for (int o = 16; o >= 1; o >>= 1) c += __shfl_xor(c, o, 32);
    if ((tid & 31) == 0) ci[tid >> 5] = c;
    __syncthreads();
    if (tid == 0) { int t = 0; for (int w = 0; w < 32; ++w) t += ci[w]; c0s = t; }
    __syncthreads();
    const int c0 = c0s;

    float sa = 0.f, sb = 0.f;
    for (int i = tid; i < N_ROWS; i += 1024) {
        const int   lab = labels[i];
        const float cnt = (float)((lab == 0 ? c0 : (N_ROWS - c0)) - 1);
        const float li  = -(pos[i] - cnt * logf(denom[i]));
        if (lab == 0) sa += li; else sb += li;
    }
    #pragma unroll
    for (int o = 16; o >= 1; o >>= 1) {
        sa += __shfl_xor(sa, o, 32);
        sb += __shfl_xor(sb, o, 32);
    }
    if ((tid & 31) == 0) { fa[tid >> 5] = sa; fb[tid >> 5] = sb; }
    __syncthreads();
    if (tid == 0) {
        float A = 0.f, B = 0.f;
        for (int w = 0; w < 32; ++w) { A += fa[w]; B += fb[w]; }
        out[0] = A / (float)c0 + B;
    }
}

extern "C" void kernel_launch(void* const* d_in, const int* in_sizes, int n_in,
                              void* d_out, int out_size, void* d_ws, size_t ws_size,
                              hipStream_t stream) {
    const float* emb    = (const float*)d_in[0];
    const int*   labels = (const int*)d_in[1];

    float* ws    = (float*)d_ws;
    float* enorm = ws;                              // N*D floats (4 MB)
    float* denom = ws + (size_t)N_ROWS * DIM;       // N floats
    float* pos   = denom + N_ROWS;                  // N floats

    scl_normalize<<<N_ROWS, DIM, 0, stream>>>(emb, enorm);
    scl_sim_rows<<<N_ROWS / 16, 256, 0, stream>>>(enorm, labels, denom, pos);
    scl_finalize<<<1, 1024, 0, stream>>>(labels, denom, pos, (float*)d_out);
}